// SDFLoss_35141422416216
// MI455X (gfx1250) — compile-verified
//
#include <hip/hip_runtime.h>
#include <hip/hip_bf16.h>
#include <math.h>
#include <stdint.h>

#define BB 8
#define NN 65536
#define GG 160
#define GG2 (GG*GG)
#define GG3 (GG*GG*GG)

typedef __attribute__((ext_vector_type(2))) float v2f;
typedef __attribute__((ext_vector_type(8))) float v8f;

// ---------------- Kernel A: SE3 exp + compose, writes se3 (B x 4 x 4) ----------------
__global__ void se3_kernel(const float* __restrict__ delta,
                           const float* __restrict__ init,
                           float* __restrict__ se3_out) {
    int b = threadIdx.x;
    if (b >= BB) return;
    const float* d = delta + b * 6;
    float u[3] = {d[0], d[1], d[2]};
    float w[3] = {d[3], d[4], d[5]};
    float t2 = w[0]*w[0] + w[1]*w[1] + w[2]*w[2];
    float theta = sqrtf(t2 + 1e-12f);
    float A  = sinf(theta) / theta;
    float Bc = (1.0f - cosf(theta)) / (theta * theta);
    float C  = (1.0f - A) / (theta * theta);
    float W[3][3] = {{0.0f, -w[2],  w[1]},
                     { w[2], 0.0f, -w[0]},
                     {-w[1], w[0],  0.0f}};
    float W2[3][3];
    #pragma unroll
    for (int i = 0; i < 3; i++)
        #pragma unroll
        for (int j = 0; j < 3; j++) {
            float s = 0.0f;
            #pragma unroll
            for (int k = 0; k < 3; k++) s += W[i][k] * W[k][j];
            W2[i][j] = s;
        }
    float T[4][4];
    #pragma unroll
    for (int i = 0; i < 4; i++)
        #pragma unroll
        for (int j = 0; j < 4; j++) T[i][j] = (i == j) ? 1.0f : 0.0f;
    float V[3][3];
    #pragma unroll
    for (int i = 0; i < 3; i++)
        #pragma unroll
        for (int j = 0; j < 3; j++) {
            float I = (i == j) ? 1.0f : 0.0f;
            T[i][j] = I + A * W[i][j] + Bc * W2[i][j];   // R
            V[i][j] = I + Bc * W[i][j] + C * W2[i][j];
        }
    #pragma unroll
    for (int i = 0; i < 3; i++)
        T[i][3] = V[i][0]*u[0] + V[i][1]*u[1] + V[i][2]*u[2];

    const float* P = init + b * 16;
    float* O = se3_out + b * 16;
    #pragma unroll
    for (int i = 0; i < 4; i++)
        #pragma unroll
        for (int j = 0; j < 4; j++) {
            float s = 0.0f;
            #pragma unroll
            for (int k = 0; k < 4; k++) s += T[i][k] * P[k*4 + j];
            O[i*4 + j] = s;
        }
}

// ---------------- Kernel B: async-LDS point staging + WMMA transform + trilinear ----------------
// 256 threads = 8 waves. Block tile = 2048 points (24 KB) async-copied to LDS with
// global_load_async_to_lds_b128, then per wave: 8 iters x 32 points
// (2 x v_wmma_f32_16x16x4_f32 per iter). 32 blocks/batch x 8 batches = 256 blocks.
__global__ void __launch_bounds__(256)
sdf_kernel(const float* __restrict__ grids,
           const float* __restrict__ limits,
           const float* __restrict__ points,
           const float* __restrict__ se3_all,
           float* __restrict__ sdf_out,
           float* __restrict__ partials) {
    __shared__ float pts_s[2048 * 3];   // 24 KB staged point tile
    __shared__ float qsh[8 * 96];       // per-wave transpose staging: 32 pts x 3 coords
    __shared__ float red[256];

    const int tid  = threadIdx.x;
    const int lane = tid & 31;
    const int wv   = tid >> 5;
    const int b    = blockIdx.x >> 5;    // 32 blocks per batch
    const int tile = blockIdx.x & 31;

    const float* P = points + (size_t)b * NN * 3;

    // ---- async copy of the whole point tile (contiguous, fully coalesced b128) ----
    {
        const float* gsrc = P + (size_t)tile * (2048 * 3);
        uint32_t lbase = (uint32_t)(uintptr_t)(&pts_s[0]);   // LDS byte offset
        #pragma unroll
        for (int k = 0; k < 6; k++) {                        // 256 thr * 6 * 16B = 24 KB
            uint32_t off  = (uint32_t)(tid + k * 256) * 16u;
            uint32_t ldst = lbase + off;
            asm volatile("global_load_async_to_lds_b128 %0, %1, %2"
                         :: "v"(ldst), "v"(off), "s"(gsrc) : "memory");
        }
        asm volatile("s_wait_asynccnt 0x0" ::: "memory");
    }

    // ---- B-matrix for WMMA: B[k][n] = se3[n][k]  (rotation + translation) ----
    // lanes 0-15: N=lane, VGPR0->K=0, VGPR1->K=1 ; lanes 16-31: N=lane-16, K=2/3
    const float* M = se3_all + b * 16;
    const int ncol = lane & 15;
    const int krow = (lane < 16) ? 0 : 2;
    v2f Bm;
    Bm.x = (ncol < 3) ? M[ncol * 4 + krow]     : 0.0f;
    Bm.y = (ncol < 3) ? M[ncol * 4 + krow + 1] : 0.0f;

    // ---- per-batch grid limits ----
    const float* L = limits + b * 9;
    const float lo0 = L[0], lo1 = L[1], lo2 = L[2];
    const float sc0 = (float)(GG - 1) / (L[3] - lo0);
    const float sc1 = (float)(GG - 1) / (L[4] - lo1);
    const float sc2 = (float)(GG - 1) / (L[5] - lo2);
    const float dmax = (float)(GG - 1);
    const float cmax = (float)(GG - 2);

    const float* gbase = grids  + (size_t)b * GG3;
    float* sout        = sdf_out + (size_t)b * NN;
    float* qw          = qsh + wv * 96;

    __syncthreads();   // staged tile visible to all waves

    float acc = 0.0f;

    #pragma unroll 1
    for (int it = 0; it < 8; it++) {
        const int li = wv * 256 + it * 32;        // local tile index of this wave's 32 pts
        const int n0 = tile * 2048 + li;          // global point index base

        // ---- A fragments from LDS: 16 points x [x,y,z,1] (ISA 16x4 f32 layout) ----
        v2f A1, A2;
        if (lane < 16) {
            int i1 = (li + lane) * 3;
            A1.x = pts_s[i1 + 0]; A1.y = pts_s[i1 + 1];
            int i2 = (li + 16 + lane) * 3;
            A2.x = pts_s[i2 + 0]; A2.y = pts_s[i2 + 1];
        } else {
            int i1 = (li + lane - 16) * 3;
            A1.x = pts_s[i1 + 2]; A1.y = 1.0f;
            int i2 = (li + lane) * 3;
            A2.x = pts_s[i2 + 2]; A2.y = 1.0f;
        }

        v8f Cz = {};
        v8f D1 = __builtin_amdgcn_wmma_f32_16x16x4_f32(false, A1, false, Bm,
                                                       (short)0, Cz, false, false);
        v8f D2 = __builtin_amdgcn_wmma_f32_16x16x4_f32(false, A2, false, Bm,
                                                       (short)0, Cz, false, false);

        // ---- transpose D (coords in lane-columns) -> point-per-lane via LDS ----
        const int pb = (lane < 16) ? 0 : 8;
        if (ncol < 3) {
            #pragma unroll
            for (int r = 0; r < 8; r++) qw[(pb + r) * 3 + ncol]      = D1[r];
            #pragma unroll
            for (int r = 0; r < 8; r++) qw[(16 + pb + r) * 3 + ncol] = D2[r];
        }
        __syncthreads();
        const float qx = qw[lane * 3 + 0];
        const float qy = qw[lane * 3 + 1];
        const float qz = qw[lane * 3 + 2];
        __syncthreads();

        // ---- trilinear interpolation (8 taps, z-pairs contiguous, L2-resident grid) ----
        float cx = fminf(fmaxf((qx - lo0) * sc0, 0.0f), dmax);
        float cy = fminf(fmaxf((qy - lo1) * sc1, 0.0f), dmax);
        float cz = fminf(fmaxf((qz - lo2) * sc2, 0.0f), dmax);
        float c0x = fminf(floorf(cx), cmax);
        float c0y = fminf(floorf(cy), cmax);
        float c0z = fminf(floorf(cz), cmax);
        float fx = cx - c0x, fy = cy - c0y, fz = cz - c0z;
        int ix = (int)c0x, iy = (int)c0y, iz = (int)c0z;

        const float* g = gbase + (size_t)ix * GG2 + iy * GG + iz;
        float g000 = g[0],        g001 = g[1];
        float g010 = g[GG],       g011 = g[GG + 1];
        float g100 = g[GG2],      g101 = g[GG2 + 1];
        float g110 = g[GG2 + GG], g111 = g[GG2 + GG + 1];

        float ox = 1.0f - fx, oy = 1.0f - fy, oz = 1.0f - fz;
        float v = g000*ox*oy*oz + g100*fx*oy*oz + g010*ox*fy*oz + g001*ox*oy*fz
                + g110*fx*fy*oz + g101*fx*oy*fz + g011*ox*fy*fz + g111*fx*fy*fz;

        sout[n0 + lane] = v;
        acc += v * v;
    }

    // ---- deterministic block reduction of sum(v^2) ----
    red[tid] = acc;
    __syncthreads();
    #pragma unroll
    for (int off = 128; off > 0; off >>= 1) {
        if (tid < off) red[tid] += red[tid + off];
        __syncthreads();
    }
    if (tid == 0) partials[blockIdx.x] = red[0];
}

// ---------------- Kernel C: final fixed-order reduction ----------------
__global__ void reduce_kernel(const float* __restrict__ partials,
                              const float* __restrict__ delta,
                              const float* __restrict__ reg,
                              float* __restrict__ loss) {
    __shared__ float s[256];
    int t = threadIdx.x;
    s[t] = partials[t];
    __syncthreads();
    #pragma unroll
    for (int off = 128; off > 0; off >>= 1) {
        if (t < off) s[t] += s[t + off];
        __syncthreads();
    }
    if (t == 0) {
        float r = 0.0f;
        for (int i = 0; i < BB * 6; i++) { float d = delta[i]; r += d * d; }
        loss[0] = 0.5f * s[0] + reg[0] * r;
    }
}

extern "C" void kernel_launch(void* const* d_in, const int* in_sizes, int n_in,
                              void* d_out, int out_size, void* d_ws, size_t ws_size,
                              hipStream_t stream) {
    const float* pose_delta = (const float*)d_in[0];   // B*6
    const float* pose_init  = (const float*)d_in[1];   // B*16
    const float* sdf_grids  = (const float*)d_in[2];   // B*G^3
    const float* sdf_limits = (const float*)d_in[3];   // B*9
    const float* points     = (const float*)d_in[4];   // B*N*3
    const float* reg        = (const float*)d_in[5];   // 1

    float* out  = (float*)d_out;
    float* loss = out;                    // [0]
    float* sdf  = out + 1;                // [1 .. 1+B*N)
    float* se3  = out + 1 + BB * NN;      // [.. +B*16)
    float* partials = (float*)d_ws;       // 256 floats

    se3_kernel<<<1, 64, 0, stream>>>(pose_delta, pose_init, se3);
    sdf_kernel<<<256, 256, 0, stream>>>(sdf_grids, sdf_limits, points, se3, sdf, partials);
    reduce_kernel<<<1, 256, 0, stream>>>(partials, pose_delta, reg, loss);
}